// GATModel_78984448573871
// MI455X (gfx1250) — compile-verified
//
#include <hip/hip_runtime.h>
#include <hip/hip_bf16.h>

// ---------------------------------------------------------------------------
// 2-layer GAT on MI455X (gfx1250, wave32).
// GEMMs use V_WMMA_F32_16X16X4_F32 (full fp32, matches reference numerics).
// Graph softmax/aggregation via L2-resident float atomics (accumulators fit
// easily in the 192MB L2, so the 150M atomic adds resolve at L2 throughput).
// ---------------------------------------------------------------------------

#define N_NODES   50000
#define N_EDGES   800000
#define IN_DIM    128
#define HID       32
#define HEADS     4
#define OUT_DIM   64
#define NEG_SLOPE 0.2f

typedef float v2f __attribute__((ext_vector_type(2)));
typedef float v8f __attribute__((ext_vector_type(8)));

// ---------------------------------------------------------------------------
// C[M x Nc] = A[M x K] * B[K x Nc], fp32, via v_wmma_f32_16x16x4_f32.
// One 16x16 C tile per wave32; 8 waves per 256-thread block.
// Requires M%16==0, K%4==0, Nc%16==0 (holds: 50000/16=3125, K=128, Nc=128/64).
// A layout (ISA 7.12.2, 32-bit A 16x4): lanes 0..15 = rows, VGPR0/1 = K,K+1;
// lanes 16..31 = same rows, K+2,K+3.  B mirrored with lanes = columns.
// Tile guard is wave-uniform -> EXEC is all ones at every WMMA (ISA req).
// ---------------------------------------------------------------------------
__global__ __launch_bounds__(256) void gemm_f32_wmma(const float* __restrict__ A,
                                                     const float* __restrict__ B,
                                                     float* __restrict__ C,
                                                     int M, int K, int Nc) {
    const int lane = threadIdx.x & 31;
    const int wave = threadIdx.x >> 5;
    const int l15  = lane & 15;
    const int hi   = lane >> 4;                  // 0 or 1 (K sub-pair select)

    const int nColTiles = Nc >> 4;
    const int nRowTiles = M >> 4;
    const int tile = blockIdx.x * 8 + wave;      // wave-uniform
    if (tile >= nRowTiles * nColTiles) return;   // whole wave exits together
    const int tr = tile / nColTiles;
    const int tc = tile - tr * nColTiles;
    const int row0 = tr << 4;
    const int col0 = tc << 4;

    v8f acc = {};

    // Per-lane base pointers.
    const float* aRow = A + (size_t)(row0 + l15) * (size_t)K + (hi << 1);
    const float* bCol = B + (size_t)(hi << 1) * (size_t)Nc + (col0 + l15);

#pragma unroll 8
    for (int k = 0; k < K; k += 4) {
        v2f a = *(const v2f*)(aRow + k);                 // A[row][k+2*hi .. +1]
        v2f b;
        b.x = bCol[(size_t)(k    ) * Nc];                // B[k+2*hi  ][col]
        b.y = bCol[(size_t)(k + 1) * Nc];                // B[k+2*hi+1][col]
        acc = __builtin_amdgcn_wmma_f32_16x16x4_f32(
            /*neg_a=*/false, a, /*neg_b=*/false, b,
            /*c_mod=*/(short)0, acc, /*reuse_a=*/false, /*reuse_b=*/false);
    }

    // C/D layout: VGPR r -> M = r (lanes 0..15) or r+8 (lanes 16..31), N = l15.
    float* cBase = C + (size_t)(row0 + (hi << 3)) * (size_t)Nc + col0 + l15;
#pragma unroll
    for (int r = 0; r < 8; ++r)
        cBase[(size_t)r * Nc] = acc[r];
}

// ---------------------------------------------------------------------------
// el[n,h] = sum_d feat[n,h,d]*attn_l[h,d];  er likewise.  One thread per (n,h).
// ---------------------------------------------------------------------------
__global__ void attn_scores(const float* __restrict__ feat,
                            const float* __restrict__ al,
                            const float* __restrict__ ar,
                            float* __restrict__ el, float* __restrict__ er,
                            int N, int H, int D) {
    int i = blockIdx.x * blockDim.x + threadIdx.x;
    if (i >= N * H) return;
    int h = i % H;
    const float* f = feat + (size_t)i * D;
    const float* L = al + (size_t)h * D;
    const float* R = ar + (size_t)h * D;
    float sl = 0.f, sr = 0.f;
    for (int d = 0; d < D; ++d) {
        float v = f[d];
        sl += v * L[d];
        sr += v * R[d];
    }
    el[i] = sl;
    er[i] = sr;
}

// segment-max init (-inf) and segment-sum init (0)
__global__ void init_ms(float* __restrict__ m, float* __restrict__ s, int n) {
    int i = blockIdx.x * blockDim.x + threadIdx.x;
    if (i < n) { m[i] = -__builtin_huge_valf(); s[i] = 0.f; }
}

// acc[n*C + c] = bias[c]
__global__ void init_bias(float* __restrict__ acc, const float* __restrict__ bias,
                          int N, int C) {
    int i = blockIdx.x * blockDim.x + threadIdx.x;
    if (i < N * C) acc[i] = bias[i % C];
}

// Sign-aware float atomic max (correct for mixed signs given -inf init):
//  - positives ordered identically as signed ints
//  - negatives: larger float = smaller unsigned pattern -> atomicMin on uint
__device__ __forceinline__ void atomicMaxF(float* addr, float val) {
    if (val >= 0.f) atomicMax((int*)addr, __float_as_int(val));
    else            atomicMin((unsigned int*)addr, __float_as_uint(val));
}

// pass 1: e = leaky_relu(el[src]+er[dst]); stash e; segment max into m[dst,h]
__global__ void edge_max(const int* __restrict__ src, const int* __restrict__ dst,
                         const float* __restrict__ el, const float* __restrict__ er,
                         float* __restrict__ ebuf, float* __restrict__ m,
                         int E, int H) {
    int i = blockIdx.x * blockDim.x + threadIdx.x;
    if (i >= E * H) return;
    int e = i / H;
    int h = i - e * H;
    float v = el[src[e] * H + h] + er[dst[e] * H + h];
    v = (v > 0.f) ? v : v * NEG_SLOPE;
    ebuf[i] = v;
    atomicMaxF(&m[dst[e] * H + h], v);
}

// pass 2: ex = exp(e - m[dst]); stash ex; segment sum into s[dst,h]
__global__ void edge_expsum(const int* __restrict__ dst,
                            const float* __restrict__ m,
                            float* __restrict__ ebuf, float* __restrict__ s,
                            int E, int H) {
    int i = blockIdx.x * blockDim.x + threadIdx.x;
    if (i >= E * H) return;
    int e = i / H;
    int h = i - e * H;
    float ex = __expf(ebuf[i] - m[dst[e] * H + h]);
    ebuf[i] = ex;
    atomicAdd(&s[dst[e] * H + h], ex);
}

// pass 3: acc[dst,h,d] += (ex/s[dst,h]) * feat[src,h,d]; one thread per (e,h,d)
__global__ void edge_scatter(const int* __restrict__ src, const int* __restrict__ dst,
                             const float* __restrict__ ebuf, const float* __restrict__ s,
                             const float* __restrict__ feat, float* __restrict__ acc,
                             int E, int H, int D) {
    int i = blockIdx.x * blockDim.x + threadIdx.x;   // < E*H*D (<= 102.4M, fits int)
    int HD = H * D;
    if (i >= E * HD) return;
    int e = i / HD;
    int r = i - e * HD;
    int h = r / D;
    int sN = src[e], dN = dst[e];
    float alpha = ebuf[e * H + h] / s[dN * H + h];
    atomicAdd(&acc[(size_t)dN * HD + r], alpha * feat[(size_t)sN * HD + r]);
}

// ELU in place (alpha=1): x>0 ? x : exp(x)-1
__global__ void elu_inplace(float* __restrict__ x, int n) {
    int i = blockIdx.x * blockDim.x + threadIdx.x;
    if (i < n) {
        float v = x[i];
        x[i] = (v > 0.f) ? v : (__expf(v) - 1.f);
    }
}

static inline int cdiv(int a, int b) { return (a + b - 1) / b; }

extern "C" void kernel_launch(void* const* d_in, const int* in_sizes, int n_in,
                              void* d_out, int out_size, void* d_ws, size_t ws_size,
                              hipStream_t stream) {
    (void)in_sizes; (void)n_in; (void)out_size; (void)ws_size;

    const float* features = (const float*)d_in[0];
    const int*   src      = (const int*)  d_in[1];
    const int*   dst      = (const int*)  d_in[2];
    const float* W1       = (const float*)d_in[3];
    const float* al1      = (const float*)d_in[4];
    const float* ar1      = (const float*)d_in[5];
    const float* b1       = (const float*)d_in[6];
    const float* W2       = (const float*)d_in[7];
    const float* al2      = (const float*)d_in[8];
    const float* ar2      = (const float*)d_in[9];
    const float* b2       = (const float*)d_in[10];
    float*       out      = (float*)d_out;

    const int N = N_NODES, E = N_EDGES;
    const int F1 = HEADS * HID;          // 128

    // Workspace layout (fp32), ~84 MB total.
    float* p     = (float*)d_ws;
    float* feat1 = p;                p += (size_t)N * F1;      // [N,128]
    float* acc1  = p;                p += (size_t)N * F1;      // [N,128] -> h1 after ELU
    float* feat2 = p;                p += (size_t)N * OUT_DIM; // [N,64]
    float* el1   = p;                p += (size_t)N * HEADS;
    float* er1   = p;                p += (size_t)N * HEADS;
    float* m1    = p;                p += (size_t)N * HEADS;
    float* s1    = p;                p += (size_t)N * HEADS;
    float* el2   = p;                p += (size_t)N;
    float* er2   = p;                p += (size_t)N;
    float* m2    = p;                p += (size_t)N;
    float* s2    = p;                p += (size_t)N;
    float* e1    = p;                p += (size_t)E * HEADS;   // [E,4]
    float* e2    = p;                /* [E] */

    const int TB = 256;

    // ---------------- Layer 1 ----------------
    {
        int tiles = (N >> 4) * (F1 >> 4);                     // 3125*8
        gemm_f32_wmma<<<cdiv(tiles, 8), TB, 0, stream>>>(features, W1, feat1, N, IN_DIM, F1);

        attn_scores<<<cdiv(N * HEADS, TB), TB, 0, stream>>>(feat1, al1, ar1, el1, er1, N, HEADS, HID);

        init_ms<<<cdiv(N * HEADS, TB), TB, 0, stream>>>(m1, s1, N * HEADS);
        init_bias<<<cdiv(N * F1, TB), TB, 0, stream>>>(acc1, b1, N, F1);

        edge_max   <<<cdiv(E * HEADS, TB), TB, 0, stream>>>(src, dst, el1, er1, e1, m1, E, HEADS);
        edge_expsum<<<cdiv(E * HEADS, TB), TB, 0, stream>>>(dst, m1, e1, s1, E, HEADS);
        edge_scatter<<<cdiv(E * HEADS * HID, TB), TB, 0, stream>>>(src, dst, e1, s1, feat1, acc1,
                                                                   E, HEADS, HID);
        elu_inplace<<<cdiv(N * F1, TB), TB, 0, stream>>>(acc1, N * F1);   // acc1 is now h1
    }

    // ---------------- Layer 2 (H=1, D=64); mean over 1 head == identity ----
    {
        int tiles = (N >> 4) * (OUT_DIM >> 4);                // 3125*4
        gemm_f32_wmma<<<cdiv(tiles, 8), TB, 0, stream>>>(acc1, W2, feat2, N, F1, OUT_DIM);

        attn_scores<<<cdiv(N, TB), TB, 0, stream>>>(feat2, al2, ar2, el2, er2, N, 1, OUT_DIM);

        init_ms<<<cdiv(N, TB), TB, 0, stream>>>(m2, s2, N);
        init_bias<<<cdiv(N * OUT_DIM, TB), TB, 0, stream>>>(out, b2, N, OUT_DIM);

        edge_max   <<<cdiv(E, TB), TB, 0, stream>>>(src, dst, el2, er2, e2, m2, E, 1);
        edge_expsum<<<cdiv(E, TB), TB, 0, stream>>>(dst, m2, e2, s2, E, 1);
        edge_scatter<<<cdiv(E * OUT_DIM, TB), TB, 0, stream>>>(src, dst, e2, s2, feat2, out,
                                                               E, 1, OUT_DIM);
    }
}